// mulGAT_17609365914383
// MI455X (gfx1250) — compile-verified
//
#include <hip/hip_runtime.h>
#include <math.h>

#define GAT_ALPHA 0.2f
#define NFEAT 256
#define HID 64
#define NHEADS 4
#define OUTW (NHEADS * HID)   // 256

typedef float v2f __attribute__((ext_vector_type(2)));
typedef float v8f __attribute__((ext_vector_type(8)));

// ---------------------------------------------------------------------------
// Phase 1: h_all[n, 256] = x[n, 256] @ Wcat[256, 256]   (fp32 WMMA 16x16x4)
// block = 128 threads = 4 waves; block b covers rows [16b, 16b+16);
// wave w computes cols [64w, 64w+64) as four 16x16 WMMA tiles.
// ---------------------------------------------------------------------------
__global__ __launch_bounds__(128)
void gat_gemm_kernel(const float* __restrict__ X, const float* __restrict__ W,
                     float* __restrict__ H, int n_nodes) {
    const int wave = threadIdx.x >> 5;
    const int lane = threadIdx.x & 31;
    const int row0 = blockIdx.x * 16;
    const int col0 = wave * 64;

    const int mrow  = lane & 15;        // M (and N) index owned by this lane
    const int khalf = (lane >> 4) << 1; // 0 for lanes 0-15, 2 for lanes 16-31

    // clamp row for A loads so padding rows never read OOB (stores are guarded)
    int arow = row0 + mrow;
    if (arow > n_nodes - 1) arow = n_nodes - 1;
    const float* __restrict__ xrow = X + (size_t)arow * NFEAT;

    v8f acc0 = {}, acc1 = {}, acc2 = {}, acc3 = {};

    for (int k = 0; k < NFEAT; k += 4) {
        v2f afrag;
        afrag.x = xrow[k + khalf];
        afrag.y = xrow[k + khalf + 1];

#pragma unroll
        for (int t = 0; t < 4; ++t) {
            const int col  = col0 + t * 16 + mrow;       // N index for this lane
            const int head = col >> 6;
            const int j    = col & (HID - 1);
            // B(kk, col) = W[head][kk][j] ; W is [4][256][64] row-major
            const float* __restrict__ wcol = W + (size_t)head * (NFEAT * HID) + j;
            v2f bfrag;
            bfrag.x = wcol[(size_t)(k + khalf) * HID];
            bfrag.y = wcol[(size_t)(k + khalf + 1) * HID];

            v8f c = (t == 0) ? acc0 : (t == 1) ? acc1 : (t == 2) ? acc2 : acc3;
            c = __builtin_amdgcn_wmma_f32_16x16x4_f32(
                    false, afrag, false, bfrag, (short)0, c, false, false);
            if (t == 0) acc0 = c; else if (t == 1) acc1 = c;
            else if (t == 2) acc2 = c; else acc3 = c;
        }
    }

    // C/D layout: VGPR i -> row = i + 8*(lane>=16), col = lane&15
    const int rhi = (lane >> 4) * 8;
#pragma unroll
    for (int t = 0; t < 4; ++t) {
        const v8f c = (t == 0) ? acc0 : (t == 1) ? acc1 : (t == 2) ? acc2 : acc3;
        const int col = col0 + t * 16 + mrow;
#pragma unroll
        for (int i = 0; i < 8; ++i) {
            const int row = row0 + rhi + i;
            if (row < n_nodes) H[(size_t)row * OUTW + col] = c[i];
        }
    }
}

// ---------------------------------------------------------------------------
// Phase 2: alpha_src[n,h] = h_all[n, 64h:64h+64] . a[h][0:64]
//          alpha_dst[n,h] = h_all[n, 64h:64h+64] . a[h][64:128]
// one thread per (node, head), float4 vectorized.
// ---------------------------------------------------------------------------
__global__ __launch_bounds__(256)
void gat_alpha_kernel(const float* __restrict__ H, const float* __restrict__ A,
                      float* __restrict__ asrc, float* __restrict__ adst,
                      int n_nodes) {
    const int idx = blockIdx.x * blockDim.x + threadIdx.x;
    if (idx >= n_nodes * NHEADS) return;
    const int node = idx >> 2;
    const int head = idx & 3;

    const float4* __restrict__ h4  = (const float4*)(H + (size_t)node * OUTW + head * HID);
    const float4* __restrict__ a14 = (const float4*)(A + (size_t)head * (2 * HID));
    const float4* __restrict__ a24 = (const float4*)(A + (size_t)head * (2 * HID) + HID);

    float s1 = 0.f, s2 = 0.f;
#pragma unroll
    for (int j = 0; j < HID / 4; ++j) {
        const float4 hv = h4[j];
        const float4 av = a14[j];
        const float4 bv = a24[j];
        s1 += hv.x * av.x + hv.y * av.y + hv.z * av.z + hv.w * av.w;
        s2 += hv.x * bv.x + hv.y * bv.y + hv.z * bv.z + hv.w * bv.w;
    }
    asrc[idx] = s1;
    adst[idx] = s2;
}

// ---------------------------------------------------------------------------
// Phase 3: edge scatter. One wave per edge (wave32). Lane L owns the 8
// consecutive floats [8L, 8L+8) of the 256-float payload; head = L>>3.
//   e = exp(-leaky_relu(asrc[src,h] + adst[dst,h], 0.2))
//   rowsum[src,h] += e ; hp[src, :] += e * h_all[dst, :]
// hp and h_all both fit in the 192MB L2 -> atomics/gathers resolve in L2.
// ---------------------------------------------------------------------------
__global__ __launch_bounds__(256)
void gat_edge_kernel(const int* __restrict__ ei, const float* __restrict__ H,
                     const float* __restrict__ asrc, const float* __restrict__ adst,
                     float* __restrict__ rowsum, float* __restrict__ hp,
                     int n_edges) {
    const int wavesPerBlock = blockDim.x >> 5;
    const int edge = blockIdx.x * wavesPerBlock + (threadIdx.x >> 5);
    if (edge >= n_edges) return;
    const int lane = threadIdx.x & 31;

    const int src = ei[edge];
    const int dst = ei[n_edges + edge];

    const int head = lane >> 3;
    const float s  = asrc[src * NHEADS + head] + adst[dst * NHEADS + head];
    const float lr = (s > 0.f) ? s : GAT_ALPHA * s;
    const float e  = __expf(-lr);

    if ((lane & 7) == 0) {
        unsafeAtomicAdd(rowsum + src * NHEADS + head, e);
    }

    const float4* __restrict__ hd = (const float4*)(H + (size_t)dst * OUTW + lane * 8);
    float* __restrict__ o = hp + (size_t)src * OUTW + lane * 8;
    const float4 v0 = hd[0];
    const float4 v1 = hd[1];
    unsafeAtomicAdd(o + 0, e * v0.x);
    unsafeAtomicAdd(o + 1, e * v0.y);
    unsafeAtomicAdd(o + 2, e * v0.z);
    unsafeAtomicAdd(o + 3, e * v0.w);
    unsafeAtomicAdd(o + 4, e * v1.x);
    unsafeAtomicAdd(o + 5, e * v1.y);
    unsafeAtomicAdd(o + 6, e * v1.z);
    unsafeAtomicAdd(o + 7, e * v1.w);
}

// ---------------------------------------------------------------------------
// Phase 4: out = elu(elu(hp / rowsum)), with reference NaN semantics
// (rowsum==0 -> hp==0 -> 0/0=NaN -> 0). In-place on d_out.
// ---------------------------------------------------------------------------
__global__ __launch_bounds__(256)
void gat_final_kernel(float* __restrict__ out, const float* __restrict__ rowsum,
                      int n_nodes) {
    const int idx = blockIdx.x * blockDim.x + threadIdx.x;
    if (idx >= n_nodes * OUTW) return;
    const int node = idx >> 8;
    const int head = (idx & 255) >> 6;

    const float rs = rowsum[node * NHEADS + head];
    float v = out[idx];
    v = (rs != 0.f) ? (v / rs) : 0.f;
    if (!(v == v)) v = 0.f;                       // NaN guard
    const float e1 = (v  > 0.f) ? v  : expm1f(v);
    const float e2 = (e1 > 0.f) ? e1 : expm1f(e1);
    out[idx] = e2;
}

// ---------------------------------------------------------------------------
extern "C" void kernel_launch(void* const* d_in, const int* in_sizes, int n_in,
                              void* d_out, int out_size, void* d_ws, size_t ws_size,
                              hipStream_t stream) {
    const float* x  = (const float*)d_in[0];
    const int*   ei = (const int*)d_in[1];
    const float* W  = (const float*)d_in[2];
    const float* a  = (const float*)d_in[3];

    const int n_nodes = in_sizes[0] / NFEAT;
    const int n_edges = in_sizes[1] / 2;

    // workspace layout
    float* H      = (float*)d_ws;                           // n_nodes * 256
    float* asrc   = H + (size_t)n_nodes * OUTW;             // n_nodes * 4
    float* adst   = asrc + (size_t)n_nodes * NHEADS;        // n_nodes * 4
    float* rowsum = adst + (size_t)n_nodes * NHEADS;        // n_nodes * 4
    float* hp     = (float*)d_out;                          // accumulate in d_out

    // deterministic per-launch init (graph-capture safe)
    hipMemsetAsync(d_out, 0, (size_t)out_size * sizeof(float), stream);
    hipMemsetAsync(rowsum, 0, (size_t)n_nodes * NHEADS * sizeof(float), stream);

    // Phase 1: WMMA GEMM
    const int rowTiles = (n_nodes + 15) / 16;
    gat_gemm_kernel<<<rowTiles, 128, 0, stream>>>(x, W, H, n_nodes);

    // Phase 2: attention logits
    {
        const int total = n_nodes * NHEADS;
        gat_alpha_kernel<<<(total + 255) / 256, 256, 0, stream>>>(H, a, asrc, adst, n_nodes);
    }

    // Phase 3: edge scatter (1 wave per edge; 8 edges per 256-thread block)
    {
        const int edgesPerBlock = 256 / 32;
        gat_edge_kernel<<<(n_edges + edgesPerBlock - 1) / edgesPerBlock, 256, 0, stream>>>(
            ei, H, asrc, adst, rowsum, hp, n_edges);
    }

    // Phase 4: normalize + double ELU
    {
        const int total = n_nodes * OUTW;
        gat_final_kernel<<<(total + 255) / 256, 256, 0, stream>>>(hp, rowsum, n_nodes);
    }
}